// SilhouetteRasterizer_61366492725847
// MI455X (gfx1250) — compile-verified
//
#include <hip/hip_runtime.h>
#include <cstdint>

#define SIMG   224
#define KTOP   10
#define TILE   16
#define TILES  (SIMG / TILE)        // 14
#define CHUNK  256                  // faces per LDS chunk
#define RECF   24                   // floats per face record (96 B)
#define BLURF  1e-4f
#define SIGINV 1e4f                 // 1/SIGMA
#define AREPS  1e-10f

typedef float v2f __attribute__((ext_vector_type(2)));
typedef float v8f __attribute__((ext_vector_type(8)));

static __device__ __forceinline__ int imin(int a, int b) { return a < b ? a : b; }

// squared distance from p to segment [a,b]  (matches reference _edge_d2)
static __device__ __forceinline__ float seg_d2(float ax, float ay, float bx, float by,
                                               float px, float py) {
    float abx = bx - ax, aby = by - ay;
    float apx = px - ax, apy = py - ay;
    float den = fmaxf(abx * abx + aby * aby, 1e-12f);
    float t   = (apx * abx + apy * aby) / den;
    t = fminf(fmaxf(t, 0.0f), 1.0f);
    float dx = apx - t * abx, dy = apy - t * aby;
    return dx * dx + dy * dy;
}

// ---------------------------------------------------------------------------
// Per-face preprocessing: plane coefficients with 1/area folded in.
// Record (float4 x 6):
//  q0 = (A0,B0,C0,Lh0)  w0 = A0*x+B0*y+C0 ; Lh0 = |e0|^2/area^2
//  q1 = (A1,B1,C1,Lh1)
//  q2 = (A2,B2,C2,Lh2)
//  q3 = (Za,Zb,Zc,good) z = Za*x+Zb*y+Zc ; good = 1.0/0.0
//  q4 = (v0x,v0y,v1x,v1y)
//  q5 = (v2x,v2y, 0, 0)
// ---------------------------------------------------------------------------
__global__ void sil_prep_kernel(const float* __restrict__ verts,
                                const int* __restrict__ faces,
                                float* __restrict__ recs,
                                int B, int V, int F, int FP) {
    int idx = blockIdx.x * blockDim.x + threadIdx.x;
    if (idx >= B * F) return;
    int b = idx / F, f = idx - b * F;

    const int* fp = faces + ((size_t)b * F + f) * 3;
    int i0 = fp[0], i1 = fp[1], i2 = fp[2];
    const float* vb = verts + (size_t)b * V * 3;
    // fixed = vertices * [-1,-1,1]
    float v0x = -vb[i0 * 3 + 0], v0y = -vb[i0 * 3 + 1], z0 = vb[i0 * 3 + 2];
    float v1x = -vb[i1 * 3 + 0], v1y = -vb[i1 * 3 + 1], z1 = vb[i1 * 3 + 2];
    float v2x = -vb[i2 * 3 + 0], v2y = -vb[i2 * 3 + 1], z2 = vb[i2 * 3 + 2];

    float area = (v1x - v0x) * (v2y - v0y) - (v1y - v0y) * (v2x - v0x);
    bool  good = fabsf(area) > AREPS;
    float inv  = 1.0f / (good ? area : 1.0f);

    float e0x = v2x - v1x, e0y = v2y - v1y;   // for w0 (anchor v1)
    float e1x = v0x - v2x, e1y = v0y - v2y;   // for w1 (anchor v2)
    float e2x = v1x - v0x, e2y = v1y - v0y;   // for w2 (anchor v0)

    float A0 = -e0y * inv, B0 = e0x * inv, C0 = (e0y * v1x - e0x * v1y) * inv;
    float A1 = -e1y * inv, B1 = e1x * inv, C1 = (e1y * v2x - e1x * v2y) * inv;
    float A2 = -e2y * inv, B2 = e2x * inv, C2 = (e2y * v0x - e2x * v0y) * inv;

    float inv2 = inv * inv;
    float L0 = (e0x * e0x + e0y * e0y) * inv2;
    float L1 = (e1x * e1x + e1y * e1y) * inv2;
    float L2 = (e2x * e2x + e2y * e2y) * inv2;

    float Za = A0 * z0 + A1 * z1 + A2 * z2;
    float Zb = B0 * z0 + B1 * z1 + B2 * z2;
    float Zc = C0 * z0 + C1 * z1 + C2 * z2;

    float4* rp = (float4*)(recs + ((size_t)b * FP + f) * RECF);
    rp[0] = make_float4(A0, B0, C0, L0);
    rp[1] = make_float4(A1, B1, C1, L1);
    rp[2] = make_float4(A2, B2, C2, L2);
    rp[3] = make_float4(Za, Zb, Zc, good ? 1.0f : 0.0f);
    rp[4] = make_float4(v0x, v0y, v1x, v1y);
    rp[5] = make_float4(v2x, v2y, 0.0f, 0.0f);
}

__global__ void sil_zero_kernel(float* __restrict__ p, int n) {
    int i = blockIdx.x * blockDim.x + threadIdx.x;
    if (i < n) p[i] = 0.0f;
}

// ---------------------------------------------------------------------------
// Raster kernel: 1 workgroup = one 16x16 pixel tile, 1 thread = 1 pixel.
// Face records streamed into double-buffered LDS with async-to-LDS copies;
// WMMA 16x16x4 f32 evaluates edge/z planes at the tile corners for a
// conservative 16-faces-at-a-time cull before the per-pixel loop.
// ---------------------------------------------------------------------------
__global__ __launch_bounds__(256) void sil_raster_kernel(
        const int* __restrict__ faces,
        const float* __restrict__ recs,
        float* __restrict__ img, float* __restrict__ vis,
        int B, int V, int F, int FP) {
    __shared__ float4 sh[2][CHUNK * 6];            // 2 x 24 KB staging
    __shared__ unsigned int kmask[CHUNK / 16];     // per-16-face keep masks

    const int b      = blockIdx.y;
    const int tileId = blockIdx.x;
    const int tx = tileId % TILES, ty = tileId / TILES;
    const int t  = threadIdx.x;
    const int ix = t & (TILE - 1), iy = t >> 4;
    const int gx = tx * TILE + ix, gy = ty * TILE + iy;
    const float px = 1.0f - (2.0f * gx + 1.0f) / (float)SIMG;
    const float py = 1.0f - (2.0f * gy + 1.0f) / (float)SIMG;

    const unsigned long long recBase =
        (unsigned long long)(size_t)(recs + (size_t)b * FP * RECF);
    const int nChunks = FP / CHUNK;

    // ---- WMMA A operand: tile-corner sample points, rows [x,y,1,0] ----
    const int  lane   = t & 31;
    const int  mrow   = lane & 15;
    const bool hiHalf = lane >= 16;
    v2f aop;
    {
        float cx0 = 1.0f - (2.0f * (tx * TILE + TILE - 1) + 1.0f) / (float)SIMG;
        float cx1 = 1.0f - (2.0f * (tx * TILE) + 1.0f) / (float)SIMG;
        float cy0 = 1.0f - (2.0f * (ty * TILE + TILE - 1) + 1.0f) / (float)SIMG;
        float cy1 = 1.0f - (2.0f * (ty * TILE) + 1.0f) / (float)SIMG;
        float ax = 0.0f, ay = 0.0f;
        if      (mrow == 0) { ax = cx0; ay = cy0; }
        else if (mrow == 1) { ax = cx1; ay = cy0; }
        else if (mrow == 2) { ax = cx0; ay = cy1; }
        else if (mrow == 3) { ax = cx1; ay = cy1; }
        if (hiHalf) { ax = 1.0f; ay = 0.0f; }      // K=2 -> 1, K=3 -> 0
        aop.x = ax; aop.y = ay;
    }

    // ---- per-pixel top-K (z ascending) ----
    float zl[KTOP], sl[KTOP];
    int   il[KTOP];
#pragma unroll
    for (int k = 0; k < KTOP; ++k) { zl[k] = 3.0e38f; sl[k] = 0.0f; il[k] = -1; }

    // ---- prologue: async-copy chunk 0 into buffer 0 ----
    {
        unsigned lds0 = (unsigned)(size_t)(&sh[0][0]);
#pragma unroll
        for (int i = 0; i < 6; ++i) {
            unsigned off = (unsigned)(i * 4096 + t * 16);
            unsigned la  = lds0 + off;
            unsigned long long ga = recBase + off;
            asm volatile("global_load_async_to_lds_b128 %0, %1, off"
                         :: "v"(la), "v"(ga) : "memory");
        }
    }

    for (int c = 0; c < nChunks; ++c) {
        const int cur = c & 1;
        if (c + 1 < nChunks) {
            unsigned lds0 = (unsigned)(size_t)(&sh[cur ^ 1][0]);
            unsigned long long gbase = recBase + (unsigned long long)(c + 1) * CHUNK * RECF * 4ull;
#pragma unroll
            for (int i = 0; i < 6; ++i) {
                unsigned off = (unsigned)(i * 4096 + t * 16);
                unsigned la  = lds0 + off;
                unsigned long long ga = gbase + off;
                asm volatile("global_load_async_to_lds_b128 %0, %1, off"
                             :: "v"(la), "v"(ga) : "memory");
            }
            asm volatile("s_wait_asynccnt 6" ::: "memory");   // chunk c done
        } else {
            asm volatile("s_wait_asynccnt 0" ::: "memory");
        }
        __syncthreads();   // chunk c resident for all waves

        const float4* R = &sh[cur][0];

        // ===== Stage A: WMMA tile cull, wave w handles groups w, w+8 =====
        const int wv = t >> 5;
#pragma unroll
        for (int g = wv; g < CHUNK / 16; g += 8) {
            const int fidx = g * 16 + mrow;
            const float4 q0 = R[fidx * 6 + 0];
            const float4 q1 = R[fidx * 6 + 1];
            const float4 q2 = R[fidx * 6 + 2];
            const float4 q3 = R[fidx * 6 + 3];
            v2f b0, b1, b2, bz;
            b0.x = hiHalf ? q0.z : q0.x;  b0.y = hiHalf ? 0.0f : q0.y;
            b1.x = hiHalf ? q1.z : q1.x;  b1.y = hiHalf ? 0.0f : q1.y;
            b2.x = hiHalf ? q2.z : q2.x;  b2.y = hiHalf ? 0.0f : q2.y;
            bz.x = hiHalf ? q3.z : q3.x;  bz.y = hiHalf ? 0.0f : q3.y;
            v8f cz = {};
            v8f d0 = __builtin_amdgcn_wmma_f32_16x16x4_f32(false, aop, false, b0, (short)0, cz, false, false);
            v8f d1 = __builtin_amdgcn_wmma_f32_16x16x4_f32(false, aop, false, b1, (short)0, cz, false, false);
            v8f d2 = __builtin_amdgcn_wmma_f32_16x16x4_f32(false, aop, false, b2, (short)0, cz, false, false);
            v8f dz = __builtin_amdgcn_wmma_f32_16x16x4_f32(false, aop, false, bz, (short)0, cz, false, false);
            bool keep = false;
            if (!hiHalf) {
                // m_i = max of the edge/z function over the 4 tile corners
                float m0 = fmaxf(fmaxf(d0[0], d0[1]), fmaxf(d0[2], d0[3]));
                float m1 = fmaxf(fmaxf(d1[0], d1[1]), fmaxf(d1[2], d1[3]));
                float m2 = fmaxf(fmaxf(d2[0], d2[1]), fmaxf(d2[2], d2[3]));
                float mz = fmaxf(fmaxf(dz[0], dz[1]), fmaxf(dz[2], dz[3]));
                // m < -sqrt(BLUR*Lhat)  <=>  m < 0 && m*m > BLUR*Lhat  (sqrt-free)
                bool cull0 = (m0 < 0.0f) & (m0 * m0 > BLURF * q0.w);
                bool cull1 = (m1 < 0.0f) & (m1 * m1 > BLURF * q1.w);
                bool cull2 = (m2 < 0.0f) & (m2 * m2 > BLURF * q2.w);
                bool cullz = !(mz > 0.0f);               // z <= 0 everywhere
                keep = (q3.w > 0.5f) && !cull0 && !cull1 && !cull2 && !cullz;
            }
            unsigned mask = __builtin_amdgcn_ballot_w32(keep);
            if (lane == 0) kmask[g] = mask & 0xFFFFu;
        }
        __syncthreads();   // kmask ready

        // ===== Stage B: per-pixel scan of kept faces =====
        const int nf = imin(CHUNK, F - c * CHUNK);
        for (int j0 = 0; j0 < nf; j0 += 16) {
            const unsigned km = kmask[j0 >> 4];
            if (km == 0u) continue;
            const int jend = imin(nf, j0 + 16);
            for (int j = j0; j < jend; ++j) {
                if (!((km >> (j - j0)) & 1u)) continue;
                const float4 q0 = R[j * 6 + 0];
                const float4 q1 = R[j * 6 + 1];
                const float4 q2 = R[j * 6 + 2];
                const float4 q3 = R[j * 6 + 3];
                float w0 = fmaf(q0.x, px, fmaf(q0.y, py, q0.z));
                float w1 = fmaf(q1.x, px, fmaf(q1.y, py, q1.z));
                float w2 = fmaf(q2.x, px, fmaf(q2.y, py, q2.z));
                float z  = fmaf(q3.x, px, fmaf(q3.y, py, q3.z));
                if (!(z > 0.0f) || !(z < zl[KTOP - 1])) continue;
                bool inside = (w0 > 0.0f) & (w1 > 0.0f) & (w2 > 0.0f);
                if (!inside) {
                    // dist(p,line_i)^2 = w_i^2 / Lhat_i ; if > BLUR on a violated
                    // edge, triangle distance^2 > BLUR -> invalid.
                    bool rej = ((w0 < 0.0f) & (w0 * w0 > BLURF * q0.w)) |
                               ((w1 < 0.0f) & (w1 * w1 > BLURF * q1.w)) |
                               ((w2 < 0.0f) & (w2 * w2 > BLURF * q2.w));
                    if (rej) continue;
                }
                const float4 q4 = R[j * 6 + 4];
                const float4 q5 = R[j * 6 + 5];
                float d2 = seg_d2(q4.x, q4.y, q4.z, q4.w, px, py);
                d2 = fminf(d2, seg_d2(q4.z, q4.w, q5.x, q5.y, px, py));
                d2 = fminf(d2, seg_d2(q5.x, q5.y, q4.x, q4.y, px, py));
                float sd;
                if (inside) sd = -d2;
                else { if (d2 > BLURF) continue; sd = d2; }
                // sorted insertion (ascending z, stable for ties -> lower idx first)
                float cz2 = z, cs = sd; int ci = c * CHUNK + j;
#pragma unroll
                for (int k = 0; k < KTOP; ++k) {
                    if (cz2 < zl[k]) {
                        float tz = zl[k], ts = sl[k]; int ti = il[k];
                        zl[k] = cz2; sl[k] = cs; il[k] = ci;
                        cz2 = tz; cs = ts; ci = ti;
                    }
                }
            }
        }
        __syncthreads();   // done reading this buffer before it is re-filled
    }

    // ---- epilogue: alpha and visibility (single divide) ----
    float pnum = 1.0f, pden = 1.0f;
#pragma unroll
    for (int k = 0; k < KTOP; ++k) {
        if (il[k] >= 0) {
            float e = __expf(sl[k] * SIGINV);   // 1-sigmoid(-sd/S) = e/(1+e)
            pnum *= e;
            pden *= 1.0f + e;
        }
    }
    img[(size_t)b * SIMG * SIMG + (size_t)gy * SIMG + gx] = 1.0f - pnum / pden;

    if (il[0] >= 0) {
        const int* fp = faces + ((size_t)b * F + il[0]) * 3;
        vis[(size_t)b * V + fp[0]] = 1.0f;
        vis[(size_t)b * V + fp[1]] = 1.0f;
        vis[(size_t)b * V + fp[2]] = 1.0f;
    }
}

// ---------------------------------------------------------------------------
extern "C" void kernel_launch(void* const* d_in, const int* in_sizes, int n_in,
                              void* d_out, int out_size, void* d_ws, size_t ws_size,
                              hipStream_t stream) {
    const float* verts = (const float*)d_in[0];   // (B,V,3) f32
    const int*   faces = (const int*)d_in[1];     // (B,F,3) i32
    const int B  = 2;
    const int V  = in_sizes[0] / (B * 3);         // 5023
    const int F  = in_sizes[1] / (B * 3);         // 5000
    const int FP = ((F + CHUNK - 1) / CHUNK) * CHUNK;   // padded to 5120

    float* recs = (float*)d_ws;                         // B*FP*24 f32 (~0.94 MB)
    float* img  = (float*)d_out;                        // B*S*S
    float* vis  = img + (size_t)B * SIMG * SIMG;        // B*V

    const int nv = B * V;
    sil_zero_kernel<<<(nv + 255) / 256, 256, 0, stream>>>(vis, nv);
    sil_prep_kernel<<<(B * F + 255) / 256, 256, 0, stream>>>(verts, faces, recs, B, V, F, FP);
    dim3 grid(TILES * TILES, B);
    sil_raster_kernel<<<grid, 256, 0, stream>>>(faces, recs, img, vis, B, V, F, FP);
}